// GloableLocalFeatureSelector_10892037062873
// MI455X (gfx1250) — compile-verified
//
#include <hip/hip_runtime.h>
#include <cstdint>

// Problem constants (from reference)
#define BB 16
#define CC 1024
#define TT 16
#define HH 12
#define WW 30
#define PP (HH * WW)     // 360 patches
#define KK 120           // top-k
#define OUTROWS 482      // 1 + 120 + 1 + 360
#define TP (TT * PP)     // 5760: stride between channels in x

typedef float v2f __attribute__((ext_vector_type(2)));
typedef float v8f __attribute__((ext_vector_type(8)));

// ---------------------------------------------------------------------------
// Kernel A: one block per batch b (384 threads = 12 wave32).
//  - compute frame-0 scores[p] = dot(cls[b,0,:], xp[b,0,p,:]) with
//    V_WMMA_F32_16X16X4_F32 (A = 16 patches x 4 channels, B = cls chunk
//    broadcast along N), accumulating 256 K-steps.
//  - stable rank-select top-120 indices (matches jax.lax.top_k ordering)
//  - copy the two cls output rows.
// ---------------------------------------------------------------------------
__global__ __launch_bounds__(384)
void glfs_scores_topk(const float* __restrict__ x,
                      const float* __restrict__ cls,
                      int* __restrict__ topk,
                      float* __restrict__ out) {
    __shared__ float clsLds[CC];        // cls token for frame 0
    __shared__ float scoresLds[368];    // 23 tiles * 16 patches (>= 360)

    const int b   = blockIdx.x;
    const int tid = threadIdx.x;
    const int wv   = tid >> 5;
    const int lane = tid & 31;

    // preload frame-0 cls token into LDS
    const float* cls0 = cls + (size_t)(b * TT + 0) * CC;
    for (int i = tid; i < CC; i += 384) clsLds[i] = cls0[i];

    // copy cls output rows (row 0 <- cls[b,0], row 121 <- cls[b,1])
    const float* cls1 = cls + (size_t)(b * TT + 1) * CC;
    float* outRow0   = out + ((size_t)b * OUTROWS + 0)   * CC;
    float* outRow121 = out + ((size_t)b * OUTROWS + 121) * CC;
    for (int i = tid; i < CC; i += 384) {
        outRow0[i]   = cls0[i];
        outRow121[i] = cls1[i];
    }
    __syncthreads();

    // ---- WMMA score computation, frame t=0 ----
    // xp(b,0,p,c) = x[(size_t)b*CC*TP + (size_t)c*TP + p]
    const float* xb_base = x + (size_t)b * CC * TP;
    const int mrow = lane & 15;          // patch row within tile
    const int ksel = (lane >> 4) * 2;    // lanes 0-15: K=0,1 ; lanes 16-31: K=2,3

    // 23 tiles of 16 patches cover 360 (last tile padded; pads never ranked)
    for (int tile = wv; tile < 23; tile += 12) {
        const int p0 = tile * 16;
        const float* xb = xb_base + (p0 + mrow);

        v8f acc = {};
#pragma unroll 4
        for (int k = 0; k < CC; k += 4) {
            const int kk = k + ksel;
            v2f a, bv;
            a[0]  = xb[(size_t)kk * TP];
            a[1]  = xb[(size_t)(kk + 1) * TP];
            bv[0] = clsLds[kk];
            bv[1] = clsLds[kk + 1];
            acc = __builtin_amdgcn_wmma_f32_16x16x4_f32(
                false, a, false, bv, (short)0, acc, false, false);
        }
        // D layout: lane 0 holds M=0..7 (N=0) in acc[0..7]; lane 16 holds M=8..15
        if (lane == 0) {
#pragma unroll
            for (int j = 0; j < 8; ++j) scoresLds[p0 + j] = acc[j];
        } else if (lane == 16) {
#pragma unroll
            for (int j = 0; j < 8; ++j) scoresLds[p0 + 8 + j] = acc[j];
        }
    }
    __syncthreads();

    // ---- stable descending rank -> top-K indices (jax.lax.top_k semantics)
    if (tid < PP) {
        const float si = scoresLds[tid];
        int r = 0;
        for (int j = 0; j < PP; ++j) {
            const float sj = scoresLds[j];
            r += (sj > si) || (sj == si && j < tid);
        }
        if (r < KK) topk[b * KK + r] = tid;
    }
}

// ---------------------------------------------------------------------------
// Kernel B: tiled [C,P] -> [P,C] transpose-copy.
//   blockDim (32,8); gridDim (32 c-tiles, 16 p-tiles, 16 b)
//   p-tiles 0..11  : frame-1 global tokens -> out rows 122..481
//   p-tiles 12..15 : frame-0 top-K gather  -> out rows 1..120
// ---------------------------------------------------------------------------
__global__ __launch_bounds__(256)
void glfs_gather(const float* __restrict__ x,
                 const int* __restrict__ topk,
                 float* __restrict__ out) {
    __shared__ float tile[32][33];

    const int b  = blockIdx.z;
    const int tp = blockIdx.y;
    const int c0 = blockIdx.x * 32;
    const int tx = threadIdx.x;
    const int ty = threadIdx.y;

    if (tp < 12) {
        // frame 1: xp(b,1,p,c) = x[b*CC*TP + c*TP + PP + p]
        const int p0 = tp * 32;
        const int p  = p0 + tx;
        const bool pv = (p < PP);
        const float* xb = x + (size_t)b * CC * TP + PP;  // t=1 offset
#pragma unroll
        for (int i = 0; i < 4; ++i) {
            const int c = c0 + ty + 8 * i;
            tile[ty + 8 * i][tx] = pv ? xb[(size_t)c * TP + p] : 0.0f;
        }
        __syncthreads();
#pragma unroll
        for (int i = 0; i < 4; ++i) {
            const int p2 = p0 + ty + 8 * i;
            if (p2 < PP)
                out[((size_t)b * OUTROWS + 122 + p2) * CC + c0 + tx] =
                    tile[tx][ty + 8 * i];
        }
    } else {
        // frame 0 top-K gather
        const int k0 = (tp - 12) * 32;
        const int k  = k0 + tx;
        const bool kv = (k < KK);
        const int gp = kv ? topk[b * KK + k] : 0;
        const float* xb = x + (size_t)b * CC * TP;       // t=0
#pragma unroll
        for (int i = 0; i < 4; ++i) {
            const int c = c0 + ty + 8 * i;
            tile[ty + 8 * i][tx] = kv ? xb[(size_t)c * TP + gp] : 0.0f;
        }
        __syncthreads();
#pragma unroll
        for (int i = 0; i < 4; ++i) {
            const int k2 = k0 + ty + 8 * i;
            if (k2 < KK)
                out[((size_t)b * OUTROWS + 1 + k2) * CC + c0 + tx] =
                    tile[tx][ty + 8 * i];
        }
    }
}

extern "C" void kernel_launch(void* const* d_in, const int* in_sizes, int n_in,
                              void* d_out, int out_size, void* d_ws, size_t ws_size,
                              hipStream_t stream) {
    (void)in_sizes; (void)n_in; (void)out_size; (void)ws_size;
    const float* x   = (const float*)d_in[0];   // [16,1024,16,12,30] f32
    const float* cls = (const float*)d_in[1];   // [16,16,1024] f32
    float* out = (float*)d_out;                 // [16,482,1024] f32
    int* topk  = (int*)d_ws;                    // 16*120 ints

    glfs_scores_topk<<<dim3(BB), dim3(384), 0, stream>>>(x, cls, topk, out);
    glfs_gather<<<dim3(32, 16, BB), dim3(32, 8), 0, stream>>>(x, topk, out);
}